// IBGConv_74345883894225
// MI455X (gfx1250) — compile-verified
//
#include <hip/hip_runtime.h>

// Problem constants (from reference)
#define NN      50000
#define EE      400000
#define MM      450000      // EE + NN (edges + self loops), divisible by 16
#define GG      128
#define HIDC    64
#define INDIM   128
#define NCLS    10
#define EPSV    1e-5f
#define NTILES  (MM / 16)   // 28125
#define MSG_WAVES 8

typedef __attribute__((ext_vector_type(2))) float v2f;
typedef __attribute__((ext_vector_type(8))) float v8f;

// ---------------- small utility kernels ----------------

__global__ void k_fill(float* __restrict__ p, int n, float v) {
  int i = blockIdx.x * blockDim.x + threadIdx.x;
  if (i < n) p[i] = v;
}

__global__ void k_deg_accum(const int* __restrict__ ecol, const float* __restrict__ attr,
                            float* __restrict__ deg) {
  int e = blockIdx.x * blockDim.x + threadIdx.x;
  if (e < EE) atomicAdd(&deg[ecol[e]], fabsf(attr[e]));
}

__global__ void k_dinv(float* __restrict__ deg) {
  int i = blockIdx.x * blockDim.x + threadIdx.x;
  if (i < NN) {
    float d = deg[i];
    deg[i] = (d > 0.0f) ? rsqrtf(d) : 0.0f;
  }
}

__global__ void k_ew(const int* __restrict__ erow, const int* __restrict__ ecol,
                     const float* __restrict__ attr, const float* __restrict__ dinv,
                     float* __restrict__ ew) {
  int i = blockIdx.x * blockDim.x + threadIdx.x;
  if (i < MM) {
    if (i < EE) {
      ew[i] = dinv[erow[i]] * fabsf(attr[i]) * dinv[ecol[i]];
    } else {
      float di = dinv[i - EE];
      ew[i] = di * di;   // self loop weight 1.0
    }
  }
}

__global__ void k_counts(const int* __restrict__ batch, float* __restrict__ counts) {
  int i = blockIdx.x * blockDim.x + threadIdx.x;
  if (i < NN) atomicAdd(&counts[batch[i]], 1.0f);
}

// ---------------- dense GEMM: xw = Z[N x K] @ W[K x 64] (fp32 WMMA) ----------------
// grid: (NN/16, 64/16), block: 32 (one wave per 16x16 output tile)
__global__ __launch_bounds__(32) void k_gemm_xw(const float* __restrict__ Z,
                                                const float* __restrict__ W,
                                                float* __restrict__ XW, int K) {
  const int lane = threadIdx.x & 31;
  const int m    = lane & 15;          // row within tile (A) / col within tile (B,C)
  const int kb   = (lane >> 4) << 1;   // 0 or 2: K sub-offset for this lane half
  const int rowA = blockIdx.x * 16 + m;
  const int colB = blockIdx.y * 16 + m;

  v8f acc = {};
  for (int k = 0; k < K; k += 4) {
    v2f a, b;
    a.x = Z[rowA * K + k + kb];
    a.y = Z[rowA * K + k + kb + 1];
    b.x = W[(k + kb) * HIDC + colB];
    b.y = W[(k + kb + 1) * HIDC + colB];
    acc = __builtin_amdgcn_wmma_f32_16x16x4_f32(false, a, false, b, (short)0, acc,
                                                false, false);
  }
  const int rbase = blockIdx.x * 16 + ((lane >> 4) << 3);
#pragma unroll
  for (int v = 0; v < 8; ++v)
    XW[(rbase + v) * HIDC + colB] = acc[v];
}

// -------- message GEMM + scatter:  out[dst] += cat[xw[dst], xw[src], ew] @ linW + linb ----
// One wave handles one tile of 16 edges, all 64 output columns (4 WMMA col tiles).
// lin_W (129x64) + lin_b staged in LDS.
__global__ __launch_bounds__(32 * MSG_WAVES) void k_msg(
    const float* __restrict__ XW,
    const int* __restrict__ erow, const int* __restrict__ ecol,
    const float* __restrict__ ew,
    const float* __restrict__ linW,   // [129*64]
    const float* __restrict__ linb,   // [64]
    float* __restrict__ outb) {
  __shared__ float lw[129 * HIDC];
  __shared__ float lb[HIDC];
  const int tid = threadIdx.x;
  for (int i = tid; i < 129 * HIDC; i += blockDim.x) lw[i] = linW[i];
  if (tid < HIDC) lb[tid] = linb[tid];
  __syncthreads();

  const int wave = tid >> 5;
  const int lane = tid & 31;
  const int tile = blockIdx.x * MSG_WAVES + wave;
  if (tile >= NTILES) return;          // wave-uniform: EXEC stays all-ones below

  const int m  = lane & 15;
  const int kb = (lane >> 4) << 1;
  const int e  = tile * 16 + m;
  const int src = (e < EE) ? erow[e] : (e - EE);
  const int dst = (e < EE) ? ecol[e] : (e - EE);
  const float* xi = XW + (long)dst * HIDC;   // x_i  (target) -> lin_W rows 0..63
  const float* xj = XW + (long)src * HIDC;   // x_j  (source) -> lin_W rows 64..127

  v8f c[4] = {v8f{}, v8f{}, v8f{}, v8f{}};
  for (int k = 0; k < HIDC; k += 4) {
    v2f a, b;
    // x_i part
    a.x = xi[k + kb];
    a.y = xi[k + kb + 1];
#pragma unroll
    for (int ct = 0; ct < 4; ++ct) {
      b.x = lw[(k + kb) * HIDC + ct * 16 + m];
      b.y = lw[(k + kb + 1) * HIDC + ct * 16 + m];
      c[ct] = __builtin_amdgcn_wmma_f32_16x16x4_f32(false, a, false, b, (short)0,
                                                    c[ct], false, false);
    }
    // x_j part
    a.x = xj[k + kb];
    a.y = xj[k + kb + 1];
#pragma unroll
    for (int ct = 0; ct < 4; ++ct) {
      b.x = lw[(64 + k + kb) * HIDC + ct * 16 + m];
      b.y = lw[(64 + k + kb + 1) * HIDC + ct * 16 + m];
      c[ct] = __builtin_amdgcn_wmma_f32_16x16x4_f32(false, a, false, b, (short)0,
                                                    c[ct], false, false);
    }
  }

  // epilogue: + ew * lin_W[128,:] + lin_b, scatter-add into out[dst]
  const int rb = (lane >> 4) << 3;
#pragma unroll
  for (int v = 0; v < 8; ++v) {
    const int e2 = tile * 16 + rb + v;
    const int d2 = (e2 < EE) ? ecol[e2] : (e2 - EE);
    const float w2 = ew[e2];
    float* orow = outb + (long)d2 * HIDC;
#pragma unroll
    for (int ct = 0; ct < 4; ++ct) {
      const int colx = ct * 16 + m;
      float val = c[ct][v] + w2 * lw[128 * HIDC + colx] + lb[colx];
      atomicAdd(&orow[colx], val);
    }
  }
}

// ---------------- GraphNorm + pooling + classifier ----------------

__global__ void k_bias_init(float* __restrict__ outb, const float* __restrict__ bias) {
  int i = blockIdx.x * blockDim.x + threadIdx.x;
  if (i < NN * HIDC) outb[i] = bias[i & (HIDC - 1)];
}

__global__ void k_accum_mean(const float* __restrict__ outb, const int* __restrict__ batch,
                             float* __restrict__ gmean) {
  int i = blockIdx.x * blockDim.x + threadIdx.x;
  if (i < NN * HIDC) {
    int n = i >> 6, h = i & 63;
    atomicAdd(&gmean[batch[n] * HIDC + h], outb[i]);
  }
}

__global__ void k_center(float* __restrict__ outb, const int* __restrict__ batch,
                         const float* __restrict__ gmean, const float* __restrict__ counts,
                         const float* __restrict__ mscale, float* __restrict__ gvar) {
  int i = blockIdx.x * blockDim.x + threadIdx.x;
  if (i < NN * HIDC) {
    int n = i >> 6, h = i & 63, g = batch[n];
    float mean = gmean[g * HIDC + h] / counts[g];
    float cen = outb[i] - mean * mscale[h];
    outb[i] = cen;
    atomicAdd(&gvar[g * HIDC + h], cen * cen);
  }
}

__global__ void k_norm(const float* __restrict__ outb, const int* __restrict__ batch,
                       const float* __restrict__ gvar, const float* __restrict__ counts,
                       const float* __restrict__ nw, const float* __restrict__ nb,
                       float* __restrict__ z, int relu) {
  int i = blockIdx.x * blockDim.x + threadIdx.x;
  if (i < NN * HIDC) {
    int n = i >> 6, h = i & 63, g = batch[n];
    float var = gvar[g * HIDC + h] / counts[g];
    float val = nw[h] * outb[i] * rsqrtf(var + EPSV) + nb[h];
    if (relu) val = fmaxf(val, 0.0f);
    z[i] = val;
  }
}

__global__ void k_pool(const float* __restrict__ z, const int* __restrict__ batch,
                       float* __restrict__ pooled) {
  int i = blockIdx.x * blockDim.x + threadIdx.x;
  if (i < NN * HIDC) {
    int n = i >> 6, h = i & 63;
    atomicAdd(&pooled[batch[n] * HIDC + h], z[i]);
  }
}

__global__ void k_cls(const float* __restrict__ pooled, const float* __restrict__ counts,
                      const float* __restrict__ cw, const float* __restrict__ cb,
                      float* __restrict__ out) {
  int i = blockIdx.x * blockDim.x + threadIdx.x;
  if (i < GG * NCLS) {
    int g = i / NCLS, c = i % NCLS;
    float inv = 1.0f / counts[g];
    float s = 0.0f;
    for (int h = 0; h < HIDC; ++h)
      s += pooled[g * HIDC + h] * inv * cw[h * NCLS + c];
    out[i] = s + cb[c];
  }
}

// ---------------- launcher ----------------

static inline int cdiv(int a, int b) { return (a + b - 1) / b; }

extern "C" void kernel_launch(void* const* d_in, const int* in_sizes, int n_in,
                              void* d_out, int out_size, void* d_ws, size_t ws_size,
                              hipStream_t stream) {
  (void)in_sizes; (void)n_in; (void)out_size; (void)ws_size;

  const float* x     = (const float*)d_in[0];
  const int*   eidx  = (const int*)d_in[1];
  const int*   erow  = eidx;        // edge_index[0] (source j)
  const int*   ecol  = eidx + EE;   // edge_index[1] (target i)
  const float* eattr = (const float*)d_in[2];
  // d_in[3] = edge_flag (unused by reference)
  const int*   batch = (const int*)d_in[4];

  // params flattened in dict-insertion order
  const float *W[3], *linW[3], *linb[3], *bias[3], *nwt[3], *nbs[3], *nms[3];
  for (int i = 0; i < 3; ++i) {
    int b0 = 5 + 7 * i;
    W[i]    = (const float*)d_in[b0 + 0];
    linW[i] = (const float*)d_in[b0 + 1];
    linb[i] = (const float*)d_in[b0 + 2];
    bias[i] = (const float*)d_in[b0 + 3];
    nwt[i]  = (const float*)d_in[b0 + 4];
    nbs[i]  = (const float*)d_in[b0 + 5];
    nms[i]  = (const float*)d_in[b0 + 6];
  }
  const float* clsW = (const float*)d_in[26];
  const float* clsb = (const float*)d_in[27];
  float* out = (float*)d_out;

  // workspace carve-up (floats)
  float* ws     = (float*)d_ws;
  float* ew     = ws;                 // MM
  float* dinv   = ew + MM;            // NN (degree, then 1/sqrt(deg))
  float* counts = dinv + NN;          // GG
  float* xw     = counts + GG;        // NN*HIDC
  float* outb   = xw + (size_t)NN * HIDC;   // NN*HIDC
  float* zb     = outb + (size_t)NN * HIDC; // NN*HIDC
  float* gmean  = zb + (size_t)NN * HIDC;   // GG*HIDC
  float* gvar   = gmean + GG * HIDC;        // GG*HIDC
  float* pooled = gvar + GG * HIDC;         // GG*HIDC

  const int B = 256;

  // edge normalization + graph counts
  k_fill<<<cdiv(NN, B), B, 0, stream>>>(dinv, NN, 1.0f);   // self-loop degree
  k_fill<<<cdiv(GG, B), B, 0, stream>>>(counts, GG, 0.0f);
  k_deg_accum<<<cdiv(EE, B), B, 0, stream>>>(ecol, eattr, dinv);
  k_dinv<<<cdiv(NN, B), B, 0, stream>>>(dinv);
  k_ew<<<cdiv(MM, B), B, 0, stream>>>(erow, ecol, eattr, dinv, ew);
  k_counts<<<cdiv(NN, B), B, 0, stream>>>(batch, counts);

  for (int i = 0; i < 3; ++i) {
    const float* zin = (i == 0) ? x : zb;
    const int K = (i == 0) ? INDIM : HIDC;

    dim3 ggrid(NN / 16, HIDC / 16);
    k_gemm_xw<<<ggrid, 32, 0, stream>>>(zin, W[i], xw, K);

    k_bias_init<<<cdiv(NN * HIDC, B), B, 0, stream>>>(outb, bias[i]);
    k_msg<<<cdiv(NTILES, MSG_WAVES), 32 * MSG_WAVES, 0, stream>>>(
        xw, erow, ecol, ew, linW[i], linb[i], outb);

    k_fill<<<cdiv(GG * HIDC, B), B, 0, stream>>>(gmean, GG * HIDC, 0.0f);
    k_fill<<<cdiv(GG * HIDC, B), B, 0, stream>>>(gvar, GG * HIDC, 0.0f);
    k_accum_mean<<<cdiv(NN * HIDC, B), B, 0, stream>>>(outb, batch, gmean);
    k_center<<<cdiv(NN * HIDC, B), B, 0, stream>>>(outb, batch, gmean, counts, nms[i], gvar);
    k_norm<<<cdiv(NN * HIDC, B), B, 0, stream>>>(outb, batch, gvar, counts, nwt[i], nbs[i],
                                                 zb, (i != 2) ? 1 : 0);
  }

  k_fill<<<cdiv(GG * HIDC, B), B, 0, stream>>>(pooled, GG * HIDC, 0.0f);
  k_pool<<<cdiv(NN * HIDC, B), B, 0, stream>>>(zb, batch, pooled);
  k_cls<<<cdiv(GG * NCLS, B), B, 0, stream>>>(pooled, counts, clsW, clsb, out);
}